// LearnableWaveletTransform_16853451669786
// MI455X (gfx1250) — compile-verified
//
#include <hip/hip_runtime.h>

// ---------------------------------------------------------------------------
// 4-level DB4 wavedec (symmetric ext) + per-level affine, fp32.
// Memory-bound (~130 MB @ 23.3 TB/s ~= 5.6 us); structured as im2col matmuls
// on V_WMMA_F32_16X16X4_F32:  D[16r x 16c] = A[16 x 24] * H[24 x 16]
// (6 chained K=4 WMMAs; 8 cA + 8 cD columns, all 256 outputs useful).
// 16 rows per workgroup; levels 2-4 run entirely out of LDS (~228 KB < 320 KB).
// Levels are template-specialized so each tile body is branch-free, and two
// independent accumulator chains per iteration overlap WMMA latency.
// ---------------------------------------------------------------------------

typedef __attribute__((ext_vector_type(2))) float v2f;
typedef __attribute__((ext_vector_type(8))) float v8f;

#define L0LEN 4096

__constant__ float c_lo[8] = {
    -0.010597401784997278f,  0.032883011666982945f,  0.030841381835986965f,
    -0.18703481171888114f,  -0.02798376941698385f,   0.6308807679295904f,
     0.7148465705525415f,    0.23037781330885523f };
// hi[i] = lo[7-i] * (i even ? -1 : +1)
__constant__ float c_hi[8] = {
    -0.23037781330885523f,   0.7148465705525415f,   -0.6308807679295904f,
    -0.02798376941698385f,   0.18703481171888114f,   0.030841381835986965f,
    -0.032883011666982945f, -0.010597401784997278f };

// symmetric extension index map: ext[e] = x[smap(e-6)]
__device__ __forceinline__ int smap(int c, int L) {
    c = (c < 0) ? (-1 - c) : c;
    return (c < L) ? c : (2 * L - 1 - c);
}

// LDS ext-buffer pitches (Lout + 13 pad, rounded up)
#define P2 2068   // level-1 cA (2051+13=2064)
#define P3 1044   // level-2 cA (1029+13=1042)
#define P4 536    // level-3 cA ( 518+13= 531)

// ---- one 16x16 output tile: 6 fragment loads + 6 chained K=4 WMMAs --------
template<int LEVEL>
__device__ __forceinline__ v8f tile_acc(const float* __restrict__ rowp, int k0,
                                        int kb, const v2f* Bf, bool bnd)
{
    v2f af[6];
    if constexpr (LEVEL == 0) {
        const int cb = 2 * k0 + kb - 6;
        if (bnd) {
#pragma unroll
            for (int s = 0; s < 6; ++s) {
                af[s][0] = rowp[smap(cb + 4 * s,     L0LEN)];
                af[s][1] = rowp[smap(cb + 4 * s + 1, L0LEN)];
            }
        } else {
#pragma unroll
            for (int s = 0; s < 6; ++s)      // even base -> aligned b64 loads
                af[s] = *(const v2f*)&rowp[cb + 4 * s];
        }
    } else {
        const int cb = 2 * k0 + kb;          // ext fully materialized in LDS
#pragma unroll
        for (int s = 0; s < 6; ++s)
            af[s] = *(const v2f*)&rowp[cb + 4 * s];
    }
    v8f acc = {};
#pragma unroll
    for (int s = 0; s < 6; ++s)
        acc = __builtin_amdgcn_wmma_f32_16x16x4_f32(
                  false, af[s], false, Bf[s], (short)0, acc, false, false);
    return acc;
}

template<int LEVEL>
__device__ __forceinline__ void run_level(const float* __restrict__ gx,
    float* __restrict__ lds, const float* __restrict__ w,
    const float* __restrict__ b, float* __restrict__ out,
    int r0, int tid, int lane, int nn, int kb, int wave, const v2f* Bf)
{
    constexpr int LinA[4]  = { 4096, 2051, 1029, 518 };
    constexpr int Lin      = LinA[LEVEL];
    constexpr int Lout     = (Lin + 7) >> 1;       // 2051/1029/518/262
    constexpr int ntile    = Lout >> 3;            // 256/128/64/32
    constexpr int tail     = Lout & 7;             //   3/  5/ 6/ 6
    constexpr int pinA[4]  = { 0, P2, P3, P4 };
    constexpr int pin      = pinA[LEVEL];
    constexpr int poutA[4] = { P2, P3, P4, 0 };
    constexpr int pout     = poutA[LEVEL];
    constexpr int inOffA[4]  = { 0, 0, 16 * P2, 16 * (P2 + P3) };
    constexpr int outOffA[4] = { 0, 16 * P2, 16 * (P2 + P3), 0 };
    // output placement: [cA4, cD4, cD3, cD2, cD1]
    constexpr int    jD      = 4 - LEVEL;
    constexpr size_t OffDA[5] = { 0, 1073152, 2146304, 4268032, 8482816 };
    constexpr size_t OffD    = OffDA[jD];
    constexpr int    LjA[5]  = { 262, 262, 518, 1029, 2051 };
    constexpr int    LjD     = LjA[jD];

    const float* lin  = lds + inOffA[LEVEL];
    float*       lout = lds + outOffA[LEVEL];
    const float  wD = w[jD], bD = b[jD];
    const float  wA = w[0],  bA = b[0];            // used only at LEVEL 3

    const float* rowp = (LEVEL == 0) ? (gx + (size_t)nn * L0LEN)
                                     : (lin + nn * pin);
    const int rh = (lane >> 4) << 3;               // +8 rows for upper lanes

    // -------- WMMA tiles, two independent chains per iteration -------------
    for (int t0 = wave * 2; t0 < ntile; t0 += 16) {
        const int ka = t0 << 3, kbt = (t0 + 1) << 3;
        const bool bndA = (LEVEL == 0) && (t0 == 0);
        const bool bndB = (LEVEL == 0) && (t0 + 1 == ntile - 1);
        v8f accA = tile_acc<LEVEL>(rowp, ka,  kb, Bf, bndA);
        v8f accB = tile_acc<LEVEL>(rowp, kbt, kb, Bf, bndB);
#pragma unroll
        for (int half = 0; half < 2; ++half) {
            const v8f& acc = half ? accB : accA;
            const int  k0  = half ? kbt  : ka;
#pragma unroll
            for (int v = 0; v < 8; ++v) {
                const int   rl  = v + rh;
                const float val = acc[v];
                if (nn < 8) {                      // approximation columns
                    const int k = k0 + nn;
                    if constexpr (LEVEL < 3)
                        lout[rl * pout + 6 + k] = val;
                    else
                        __builtin_nontemporal_store(val * wA + bA,
                            &out[(size_t)(r0 + rl) * 262 + k]);
                } else {                           // detail columns
                    const int k = k0 + nn - 8;
                    __builtin_nontemporal_store(val * wD + bD,
                        &out[OffD + (size_t)(r0 + rl) * LjD + k]);
                }
            }
        }
    }

    // -------- ragged tail (scalar) -----------------------------------------
    const int nTail = tail * 32;                   // 16 rows * 2 filt * tail
    for (int i = tid; i < nTail; i += 256) {
        const int rl = i & 15;
        const int f  = (i >> 4) & 1;
        const int k  = (ntile << 3) + (i >> 5);
        float s = 0.0f;
#pragma unroll
        for (int t = 0; t < 8; ++t) {
            float e;
            if constexpr (LEVEL == 0)
                e = gx[(size_t)rl * L0LEN + smap(2 * k + t - 6, L0LEN)];
            else
                e = lin[rl * pin + 2 * k + t];
            s += e * (f ? c_hi[7 - t] : c_lo[7 - t]);
        }
        if (f == 0) {
            if constexpr (LEVEL < 3)
                lout[rl * pout + 6 + k] = s;
            else
                __builtin_nontemporal_store(s * wA + bA,
                    &out[(size_t)(r0 + rl) * 262 + k]);
        } else {
            __builtin_nontemporal_store(s * wD + bD,
                &out[OffD + (size_t)(r0 + rl) * LjD + k]);
        }
    }

    __syncthreads();                               // cA done before pad fill
    if constexpr (LEVEL < 3) {
        for (int i = tid; i < 16 * 13; i += 256) {
            const int rl = i / 13, p = i % 13;
            float* rp = lout + rl * pout;
            if (p < 6) rp[p] = rp[11 - p];                 // left pad
            else       rp[Lout + p] = rp[Lout + 11 - p];   // right pad
        }
    }
    __syncthreads();                               // pads visible next level
}

__global__ __launch_bounds__(256)
void dwt_wmma_kernel(const float* __restrict__ x, const float* __restrict__ w,
                     const float* __restrict__ b, float* __restrict__ out)
{
    __shared__ float lds[16 * (P2 + P3 + P4)];

    const int tid  = threadIdx.x;
    const int lane = tid & 31;
    const int nn   = lane & 15;            // A row / D column within half
    const int kb   = (lane >> 4) << 1;     // K offset {0,2} per A-layout
    const int wave = tid >> 5;
    const int r0   = blockIdx.x * 16;

    // per-lane B fragments: banded filter matrix H, six K=4 slices.
    // B layout assumed transpose-symmetric to A: lanes 0-15 -> rows K{0,1},
    // lanes 16-31 -> rows K{2,3}; column = lane&15.
    const int fsel = nn >> 3;              // 0 = low-pass cols, 1 = high-pass
    const int dk   = nn & 7;               // output offset within 8-wide tile
    v2f Bf[6];
#pragma unroll
    for (int s = 0; s < 6; ++s) {
#pragma unroll
        for (int r = 0; r < 2; ++r) {
            const int e = 4 * s + kb + r;
            const int t = e - 2 * dk;
            float v = 0.0f;
            if (t >= 0 && t < 8) v = fsel ? c_hi[7 - t] : c_lo[7 - t];
            Bf[s][r] = v;
        }
    }

    const float* gx = x + (size_t)r0 * L0LEN;
    run_level<0>(gx, lds, w, b, out, r0, tid, lane, nn, kb, wave, Bf);
    run_level<1>(gx, lds, w, b, out, r0, tid, lane, nn, kb, wave, Bf);
    run_level<2>(gx, lds, w, b, out, r0, tid, lane, nn, kb, wave, Bf);
    run_level<3>(gx, lds, w, b, out, r0, tid, lane, nn, kb, wave, Bf);
}

extern "C" void kernel_launch(void* const* d_in, const int* in_sizes, int n_in,
                              void* d_out, int out_size, void* d_ws, size_t ws_size,
                              hipStream_t stream) {
    const float* x = (const float*)d_in[0];
    const float* w = (const float*)d_in[1];
    const float* b = (const float*)d_in[2];
    float* out = (float*)d_out;
    dwt_wmma_kernel<<<256, 256, 0, stream>>>(x, w, b, out);
}